// WaveletParsingLayer_65034394796500
// MI455X (gfx1250) — compile-verified
//
#include <hip/hip_runtime.h>

// Problem constants (match reference)
#define B_DIM   128
#define N_SEGS  128
#define SEG_LEN 2048
#define K_OUT   (128 * 1024)          // 131072 outputs per row
#define FILLER  10.1f

// ---------------------------------------------------------------------------
// CDNA5 async global->LDS path (gfx1250). Guarded: falls back to plain loads.
// ---------------------------------------------------------------------------
#if defined(__HIP_DEVICE_COMPILE__) && \
    __has_builtin(__builtin_amdgcn_global_load_async_to_lds_b128) && \
    __has_builtin(__builtin_amdgcn_s_wait_asynccnt)
#define USE_ASYNC_LDS 1
#else
#define USE_ASYNC_LDS 0
#endif

// Builtin parameter types per hipcc diagnostic: v4i (gcc vector_size(16) int)
// pointers, global (as1) source and LDS (as3) destination.
typedef int v4i __attribute__((vector_size(16)));
typedef __attribute__((address_space(1))) v4i* gv4i_ptr;
typedef __attribute__((address_space(3))) v4i* lv4i_ptr;

// Wave32 inclusive scan (gfx1250 is wave32-only)
__device__ __forceinline__ unsigned wave_incl_scan(unsigned v) {
#pragma unroll
  for (int d = 1; d < 32; d <<= 1) {
    unsigned n = __shfl_up(v, d, 32);
    if ((threadIdx.x & 31u) >= (unsigned)d) v += n;
  }
  return v;
}

// ---------------------------------------------------------------------------
// Compaction kernel (placed first so its asm shows in the disasm snippet).
// One 256-thread block per (b, seg); segment staged in LDS via CDNA5 async
// global->LDS b128 copies (ASYNCcnt + s_wait_asynccnt), then stable-compacted
// with wave32 shuffle scans. Output stores are non-temporal so the 64 MiB
// write-once output does not evict the L2-resident 128 MiB x3 between passes.
// ---------------------------------------------------------------------------
__global__ __launch_bounds__(256) void compact_kernel(
    const float* __restrict__ x3, const unsigned* __restrict__ bases,
    float* __restrict__ out) {
  const unsigned blk = blockIdx.x;                 // b * N_SEGS + seg
  const unsigned row = blk / N_SEGS;

  __shared__ float tile[SEG_LEN];                  // 8 KB
  const float* gseg = x3 + (size_t)blk * SEG_LEN;

#if USE_ASYNC_LDS
  {
    // Each thread copies 16B at tid*16 and tid*16+4096 (imm offset applies to
    // both the global and LDS addresses per ISA async-load semantics).
    gv4i_ptr gp = (gv4i_ptr)(unsigned long long)(gseg + threadIdx.x * 4);
    lv4i_ptr lp = (lv4i_ptr)(unsigned)(unsigned long long)(&tile[threadIdx.x * 4]);
    __builtin_amdgcn_global_load_async_to_lds_b128(gp, lp, 0, 0);
    __builtin_amdgcn_global_load_async_to_lds_b128(gp, lp, 4096, 0);
    __builtin_amdgcn_s_wait_asynccnt(0);           // this wave's asyncs done
  }
#else
  {
    float4* lt = (float4*)tile;
    const float4* gt = (const float4*)gseg;
#pragma unroll
    for (int i = 0; i < 2; ++i) lt[threadIdx.x + i * 256] = gt[threadIdx.x + i * 256];
  }
#endif
  __syncthreads();                                 // all waves' data visible

  // Each thread owns 8 consecutive elements -> stable order preserved.
  float v[8];
  unsigned mask = 0, cnt = 0;
#pragma unroll
  for (int i = 0; i < 8; ++i) {
    v[i] = tile[threadIdx.x * 8 + i];
    if (v[i] != FILLER) { mask |= (1u << i); ++cnt; }
  }

  // Block-wide exclusive scan of per-thread counts (wave32 scans + LDS).
  __shared__ unsigned wbase[8];
  const unsigned lane = threadIdx.x & 31u, wid = threadIdx.x >> 5;
  unsigned incl = wave_incl_scan(cnt);
  if (lane == 31) wbase[wid] = incl;
  __syncthreads();
  if (wid == 0) {
    unsigned t = (lane < 8) ? wbase[lane] : 0u;
    unsigned ts = wave_incl_scan(t);
    if (lane < 8) wbase[lane] = ts - t;            // exclusive wave offsets
  }
  __syncthreads();
  const unsigned excl = (incl - cnt) + wbase[wid];

  float* __restrict__ orow = out + (size_t)row * K_OUT;
  unsigned pos = bases[blk] + excl;                // global output position
#pragma unroll
  for (int i = 0; i < 8; ++i) {
    if (mask & (1u << i)) {
      if (pos < K_OUT) __builtin_nontemporal_store(v[i], &orow[pos]);
      ++pos;
    }
  }
}

// ---------------------------------------------------------------------------
// Kernel 1: per-segment valid count. One 256-thread block per (b, seg).
// Regular (RT) loads so x3 lines land in L2 for reuse by compact_kernel.
// ---------------------------------------------------------------------------
__global__ __launch_bounds__(256) void seg_count_kernel(
    const float* __restrict__ x3, unsigned* __restrict__ counts) {
  const unsigned blk = blockIdx.x;                 // b * N_SEGS + seg
  const float4* seg = (const float4*)(x3 + (size_t)blk * SEG_LEN);

  unsigned c = 0;
#pragma unroll
  for (int i = 0; i < 2; ++i) {
    float4 v = seg[threadIdx.x + i * 256];         // contiguous b128 per wave
    c += (v.x != FILLER) + (v.y != FILLER) + (v.z != FILLER) + (v.w != FILLER);
  }
  // wave32 reduce
#pragma unroll
  for (int d = 16; d > 0; d >>= 1) c += __shfl_down(c, d, 32);

  __shared__ unsigned wsum[8];
  const unsigned lane = threadIdx.x & 31u, wid = threadIdx.x >> 5;
  if (lane == 0) wsum[wid] = c;
  __syncthreads();
  if (threadIdx.x == 0) {
    unsigned t = 0;
#pragma unroll
    for (int i = 0; i < 8; ++i) t += wsum[i];
    counts[blk] = t;
  }
}

// ---------------------------------------------------------------------------
// Kernel 2: per-row exclusive scan of the 128 segment counts. One wave per row.
// ---------------------------------------------------------------------------
__global__ __launch_bounds__(32) void row_scan_kernel(
    const unsigned* __restrict__ counts, unsigned* __restrict__ bases,
    unsigned* __restrict__ totals) {
  const unsigned row = blockIdx.x;
  const unsigned t = threadIdx.x;                  // 0..31
  const unsigned* c = counts + (size_t)row * N_SEGS;

  unsigned v[4], s = 0;
#pragma unroll
  for (int i = 0; i < 4; ++i) { v[i] = c[t * 4 + i]; s += v[i]; }

  unsigned incl = wave_incl_scan(s);
  unsigned excl = incl - s;

  unsigned* b = bases + (size_t)row * N_SEGS;
#pragma unroll
  for (int i = 0; i < 4; ++i) { b[t * 4 + i] = excl; excl += v[i]; }
  if (t == 31) totals[row] = incl;                 // total valid in row
}

// ---------------------------------------------------------------------------
// Kernel 4: fill tail [total, K_OUT) with FILLER (dropped elems are all FILLER
// by definition, so this exactly reproduces the reference's sorted tail).
// ---------------------------------------------------------------------------
__global__ __launch_bounds__(256) void tail_fill_kernel(
    const unsigned* __restrict__ totals, float* __restrict__ out) {
  const unsigned row = blockIdx.x;
  unsigned start = totals[row];
  if (start > K_OUT) start = K_OUT;
  float* __restrict__ orow = out + (size_t)row * K_OUT;
  for (unsigned i = start + threadIdx.x; i < K_OUT; i += blockDim.x)
    __builtin_nontemporal_store(FILLER, &orow[i]);
}

// ---------------------------------------------------------------------------
// Launcher
// ---------------------------------------------------------------------------
extern "C" void kernel_launch(void* const* d_in, const int* in_sizes, int n_in,
                              void* d_out, int out_size, void* d_ws, size_t ws_size,
                              hipStream_t stream) {
  (void)in_sizes; (void)n_in; (void)out_size; (void)ws_size;

  const float* x3 = (const float*)d_in[2];         // [B, N_SEGS, SEG_LEN] f32
  float* out = (float*)d_out;                      // [B, K_OUT] f32

  unsigned* counts = (unsigned*)d_ws;              // 16384 u32
  unsigned* bases  = counts + B_DIM * N_SEGS;      // 16384 u32
  unsigned* totals = bases + B_DIM * N_SEGS;       // 128 u32

  seg_count_kernel<<<B_DIM * N_SEGS, 256, 0, stream>>>(x3, counts);
  row_scan_kernel<<<B_DIM, 32, 0, stream>>>(counts, bases, totals);
  compact_kernel<<<B_DIM * N_SEGS, 256, 0, stream>>>(x3, bases, out);
  tail_fill_kernel<<<B_DIM, 256, 0, stream>>>(totals, out);
}